// FCCEncoder_51926154609189
// MI455X (gfx1250) — compile-verified
//
#include <hip/hip_runtime.h>

// ---------------------------------------------------------------------------
// Wigner-D application for l=4 and l=6 over B=2M quaternions (gfx1250).
//
// f_l(b) = exp(a*Gy) exp(b*Gx) exp(g*Gy) s_l
//        = Y(a) * J * Y(b) * J * Y(g) * s_l,   J = expm(pi/sqrt(2)*(Gx+Gy))
//
// Y(theta) is sparse (2x2 blocks over +/-m); J is a constant dense matrix
// (9x9 / 13x13, zero-padded to 16x16) applied as batched matvecs with
// V_WMMA_F32_16X16X4_F32 (fp32 WMMA: exact, and this kernel is HBM-bound
// at ~210MB traffic => ~9us floor at 23.3 TB/s, so fp32 matrix path is free).
// ---------------------------------------------------------------------------

typedef float v2f __attribute__((ext_vector_type(2)));
typedef float v8f __attribute__((ext_vector_type(8)));

__device__ inline v8f wmma4(v2f a, v2f b, v8f c) {
    // 8-arg pattern: (neg_a, A, neg_b, B, c_mod, C, reuse_a, reuse_b)
    return __builtin_amdgcn_wmma_f32_16x16x4_f32(
        false, a, false, b, (short)0, c, false, false);
}

// ---------------------------------------------------------------------------
// Setup kernel: build Gy, Gx in the real-SH basis (mirrors reference's
// Re(Q^H X Q)), store signed +/-m couplings of Gy, and compute
// J = expm(pi/sqrt2 * (Gx+Gy)) by scaling-and-squaring (Horner order 9,
// 12 squarings). 16 lanes each own one column of the 16x16 padded matrix.
//
// ws layout (floats): [0..255] J4 (16x16 row-major), [256..511] J6,
//                     [512..519] ky4[m] (m=1..4), [520..527] ky6[m] (m=1..6)
// ---------------------------------------------------------------------------
__global__ void wigner_setup(float* __restrict__ ws) {
    const int l = (blockIdx.x == 0) ? 4 : 6;
    const int N = 2 * l + 1;

    __shared__ float Qr[13][13], Qi[13][13];
    __shared__ float Ms[16][16];   // pre-scaled (Gx+Gy)*pi/sqrt2 * 2^-12
    __shared__ float P[16][17];

    const int t = threadIdx.x;

    if (t == 0) {
        for (int i = 0; i < 13; ++i)
            for (int j = 0; j < 13; ++j) { Qr[i][j] = 0.f; Qi[i][j] = 0.f; }
        const float is2 = 0.70710678118654752f;
        // Q: complex->real SH basis change (global phase (-i)^l is real +/-1
        // for even l and cancels in Q^H X Q, so it is omitted).
        for (int m = -l; m < 0; ++m) {
            Qr[l + m][l - m] = is2;          // 1/sqrt2
            Qi[l + m][l + m] = -is2;         // -i/sqrt2
        }
        Qr[l][l] = 1.f;
        for (int m = 1; m <= l; ++m) {
            float sg = (m & 1) ? -1.f : 1.f;
            Qr[l + m][l + m] = sg * is2;     // (-1)^m/sqrt2
            Qi[l + m][l - m] = sg * is2;     // i(-1)^m/sqrt2
        }

        // ky[m] = Gy[l+m][l-m]  (signed coupling for the sparse Y rotation)
        const int kyoff = (l == 4) ? 512 : 520;
        for (int m = 1; m <= l; ++m) {
            float g = 0.f;
            for (int a = 0; a < N; ++a) {
                float mm = (float)(a - l);
                g += mm * (Qi[a][l + m] * Qr[a][l - m] -
                           Qr[a][l + m] * Qi[a][l - m]);
            }
            ws[kyoff + m] = g;
        }

        // Ms = (Gx+Gy) * pi/sqrt2 * 2^-12 (padded with zeros to 16x16)
        const float scale = 2.2214414690791831f * 0.000244140625f;
        for (int i = 0; i < 16; ++i)
            for (int j = 0; j < 16; ++j) Ms[i][j] = 0.f;
        for (int u = 0; u < N; ++u) {
            for (int v = 0; v < N; ++v) {
                float gy = 0.f, gx = 0.f;
                for (int a = 0; a < N; ++a) {
                    float mm = (float)(a - l);
                    // Gy = Re(Q^H * i*diag(m) * Q)
                    gy += mm * (Qi[a][u] * Qr[a][v] - Qr[a][u] * Qi[a][v]);
                    // Gx = Re(Q^H * X0 * Q), X0 antisym tri-diagonal:
                    // X0[a+1][a] = -c_a/2, X0[a][a+1] = +c_a/2
                    float tr = 0.f, ti = 0.f;
                    if (a > 0) {
                        float am = (float)(a - 1 - l);
                        float cm1 = sqrtf((float)(l * (l + 1)) - am * (am + 1.f));
                        tr -= 0.5f * cm1 * Qr[a - 1][v];
                        ti -= 0.5f * cm1 * Qi[a - 1][v];
                    }
                    if (a < N - 1) {
                        float a0 = (float)(a - l);
                        float c0 = sqrtf((float)(l * (l + 1)) - a0 * (a0 + 1.f));
                        tr += 0.5f * c0 * Qr[a + 1][v];
                        ti += 0.5f * c0 * Qi[a + 1][v];
                    }
                    gx += Qr[a][u] * tr + Qi[a][u] * ti;
                }
                Ms[u][v] = (gx + gy) * scale;
            }
        }
    }
    __syncthreads();

    const int j = t;            // lanes 0..15 own columns
    if (j < 16)
        for (int i = 0; i < 16; ++i) P[i][j] = (i == j) ? 1.f : 0.f;
    __syncthreads();

    // Horner: P = I + Ms*(I + Ms/2*(... (I + Ms/9) ...))
    for (int k = 9; k >= 1; --k) {
        float col[16];
        if (j < 16)
            for (int i = 0; i < 16; ++i) {
                float s = 0.f;
                for (int m = 0; m < 16; ++m) s += Ms[i][m] * P[m][j];
                col[i] = s;
            }
        __syncthreads();
        if (j < 16) {
            float inv = 1.f / (float)k;
            for (int i = 0; i < 16; ++i)
                P[i][j] = ((i == j) ? 1.f : 0.f) + col[i] * inv;
        }
        __syncthreads();
    }
    // 12 squarings (undo 2^-12 scaling)
    for (int sq = 0; sq < 12; ++sq) {
        float col[16];
        if (j < 16)
            for (int i = 0; i < 16; ++i) {
                float s = 0.f;
                for (int m = 0; m < 16; ++m) s += P[i][m] * P[m][j];
                col[i] = s;
            }
        __syncthreads();
        if (j < 16)
            for (int i = 0; i < 16; ++i) P[i][j] = col[i];
        __syncthreads();
    }
    if (j < 16) {
        float* Jo = ws + ((l == 4) ? 0 : 256);
        for (int i = 0; i < 16; ++i) Jo[i * 16 + j] = P[i][j];
    }
}

// ---------------------------------------------------------------------------
// Main kernel helpers
// ---------------------------------------------------------------------------

// Sparse Y-rotation: 2x2 blocks over (l+m, l-m), angle coupling ky[m]=+/-m.
template <int L>
__device__ inline void yrot(float* v, const float* cc, const float* ss,
                            const float* ky) {
#pragma unroll
    for (int m = 1; m <= L; ++m) {
        float k = ky[m];
        float sm = (k < 0.f) ? -ss[m] : ss[m];
        float cm = cc[m];
        float hi = v[L + m], lo = v[L - m];
        v[L + m] = cm * hi + sm * lo;
        v[L - m] = cm * lo - sm * hi;
    }
}

// Dense constant matvec out = J*v for 32 quats/wave via 2 WMMA tiles.
// v (lane-major, 16 floats/lane) -> per-wave LDS -> WMMA B-layout ->
// 4-chunk K=16 chain of V_WMMA_F32_16X16X4_F32 -> scatter D -> lane-major.
// All LDS traffic is intra-wave (DS ops are in-order per wave; no barrier).
__device__ inline void j_apply(float (*row)[17], int lane, const v2f* aJ,
                               float* v) {
    const int lm = lane & 15, lh = lane >> 4;
#pragma unroll
    for (int d = 0; d < 16; ++d) row[lane][d] = v[d];

    v8f acc0 = {};
    v8f acc1 = {};
#pragma unroll
    for (int c = 0; c < 4; ++c) {
        int kk = 4 * c + 2 * lh;
        // B 4x16 layout mirrors A: VGPR0 -> K=kk rows, VGPR1 -> K=kk+1
        v2f b0, b1;
        b0.x = row[lm][kk];      b0.y = row[lm][kk + 1];        // tile 0: quats 0..15
        b1.x = row[16 + lm][kk]; b1.y = row[16 + lm][kk + 1];   // tile 1: quats 16..31
        acc0 = wmma4(aJ[c], b0, acc0);
        acc1 = wmma4(aJ[c], b1, acc1);
    }
    // D layout: vgpr r, lane c holds (M = r + 8*(c>>4), N = c&15)
#pragma unroll
    for (int r = 0; r < 8; ++r) {
        row[lm][r + 8 * lh]      = acc0[r];
        row[16 + lm][r + 8 * lh] = acc1[r];
    }
#pragma unroll
    for (int d = 0; d < 16; ++d) v[d] = row[lane][d];
}

__global__ __launch_bounds__(256) void wigner_main(
    const float* __restrict__ quats, const float* __restrict__ s4g,
    const float* __restrict__ s6g, float* __restrict__ out,
    const float* __restrict__ ws, int B) {
    __shared__ float smem[8][32][17];   // per-wave staging, 17-stride pad

    const int lane = threadIdx.x & 31;
    const int wv = threadIdx.x >> 5;
    const int lm = lane & 15, lh = lane >> 4;

    long gid = (long)blockIdx.x * 256 + threadIdx.x;
    bool active = gid < (long)B;
    long q = active ? gid : (long)(B - 1);   // tail lanes: valid data, no store
                                             // (keeps EXEC full for WMMA)

    // J matrices in WMMA A-layout: vgpr0 lanes0-15 K=4c, lanes16-31 K=4c+2
    v2f a4[4], a6[4];
    const float* J4 = ws;
    const float* J6 = ws + 256;
#pragma unroll
    for (int c = 0; c < 4; ++c) {
        int kk = 4 * c + 2 * lh;
        a4[c].x = J4[lm * 16 + kk]; a4[c].y = J4[lm * 16 + kk + 1];
        a6[c].x = J6[lm * 16 + kk]; a6[c].y = J6[lm * 16 + kk + 1];
    }
    float ky4[7] = {}, ky6[7] = {};
#pragma unroll
    for (int m = 1; m <= 4; ++m) ky4[m] = ws[512 + m];
#pragma unroll
    for (int m = 1; m <= 6; ++m) ky6[m] = ws[520 + m];

    // ---- quaternion -> cos/sin of Euler angles, transcendental-free ----
    const float4 qq = ((const float4*)quats)[q];
    float w = qq.x, x = qq.y, y = qq.z, z = qq.w;
    float R01 = 2.f * (x * y - w * z);
    float R11 = 1.f - 2.f * (x * x + z * z);
    float R21 = 2.f * (y * z + w * x);
    float R00 = 1.f - 2.f * (y * y + z * z);
    float R20 = 2.f * (x * z - w * y);
    float R02 = 2.f * (x * z + w * y);
    float R22 = 1.f - 2.f * (x * x + y * y);

    float n = sqrtf(R01 * R01 + R11 * R11 + R21 * R21);
    float innv = 1.f / fmaxf(n, 1e-12f);
    float xcx = fminf(fmaxf(R01 * innv, -1.f), 1.f);
    float xcy = fminf(fmaxf(R11 * innv, -1.f), 1.f);
    float xcz = fminf(fmaxf(R21 * innv, -1.f), 1.f);

    float cb = xcy;                                   // cos(beta)
    float sb = sqrtf(fmaxf(1.f - cb * cb, 0.f));      // sin(beta) >= 0
    float rho = sqrtf(xcx * xcx + xcz * xcz);
    float irho = 1.f / fmaxf(rho, 1e-12f);
    float ca = (rho < 1e-9f) ? 1.f : xcz * irho;      // cos(alpha)
    float sa = (rho < 1e-9f) ? 0.f : xcx * irho;      // sin(alpha)

    float M00 = ca * R00 - sa * R20;
    float M02 = ca * R02 - sa * R22;
    float rr = sqrtf(M00 * M00 + M02 * M02);
    float irr = 1.f / fmaxf(rr, 1e-12f);
    float cg = (rr < 1e-9f) ? 1.f : M00 * irr;        // cos(gamma)
    float sg = (rr < 1e-9f) ? 0.f : M02 * irr;        // sin(gamma)

    // Chebyshev: cos/sin(m*theta), m = 0..6, 2 fma each
    float cA[7], sA[7], cB[7], sB[7], cG[7], sG[7];
    cA[0] = 1.f; sA[0] = 0.f; cA[1] = ca; sA[1] = sa;
    cB[0] = 1.f; sB[0] = 0.f; cB[1] = cb; sB[1] = sb;
    cG[0] = 1.f; sG[0] = 0.f; cG[1] = cg; sG[1] = sg;
#pragma unroll
    for (int m = 2; m < 7; ++m) {
        cA[m] = 2.f * ca * cA[m - 1] - cA[m - 2];
        sA[m] = 2.f * ca * sA[m - 1] - sA[m - 2];
        cB[m] = 2.f * cb * cB[m - 1] - cB[m - 2];
        sB[m] = 2.f * cb * sB[m - 1] - sB[m - 2];
        cG[m] = 2.f * cg * cG[m - 1] - cG[m - 2];
        sG[m] = 2.f * cg * sG[m - 1] - sG[m - 2];
    }

    float v[16];

    // ----- l = 4 -----
#pragma unroll
    for (int d = 0; d < 16; ++d) v[d] = (d < 9) ? s4g[d] : 0.f;
    yrot<4>(v, cG, sG, ky4);
    j_apply(smem[wv], lane, a4, v);
    yrot<4>(v, cB, sB, ky4);
    j_apply(smem[wv], lane, a4, v);
    yrot<4>(v, cA, sA, ky4);
    if (active) {
#pragma unroll
        for (int i = 0; i < 9; ++i) out[q * 9 + i] = v[i];
    }

    // ----- l = 6 -----
#pragma unroll
    for (int d = 0; d < 16; ++d) v[d] = (d < 13) ? s6g[d] : 0.f;
    yrot<6>(v, cG, sG, ky6);
    j_apply(smem[wv], lane, a6, v);
    yrot<6>(v, cB, sB, ky6);
    j_apply(smem[wv], lane, a6, v);
    yrot<6>(v, cA, sA, ky6);
    if (active) {
        float* out6 = out + (long)B * 9;
#pragma unroll
        for (int i = 0; i < 13; ++i) out6[q * 13 + i] = v[i];
    }
}

// ---------------------------------------------------------------------------
extern "C" void kernel_launch(void* const* d_in, const int* in_sizes, int n_in,
                              void* d_out, int out_size, void* d_ws,
                              size_t ws_size, hipStream_t stream) {
    (void)n_in; (void)out_size; (void)ws_size;
    const float* quats = (const float*)d_in[0];
    const float* s4 = (const float*)d_in[1];
    const float* s6 = (const float*)d_in[2];
    float* out = (float*)d_out;
    float* ws = (float*)d_ws;
    const int B = in_sizes[0] / 4;

    wigner_setup<<<2, 32, 0, stream>>>(ws);                 // ~2KB of ws
    const int blocks = (B + 255) / 256;                     // 256 thr = 8 waves
    wigner_main<<<blocks, 256, 0, stream>>>(quats, s4, s6, out, ws, B);
}